// BEAM_LAMP_layer_2731599201053
// MI455X (gfx1250) — compile-verified
//
#include <hip/hip_runtime.h>
#include <math.h>

#define BATCH 65536
#define NN 512
#define KD 48

typedef float v2f __attribute__((ext_vector_type(2)));
typedef float v8f __attribute__((ext_vector_type(8)));

// ---------------- kernel 0: zero scalar accumulators ----------------
__global__ void lamp_init_ws(float* __restrict__ ws) {
  if (threadIdx.x < 3) ws[threadIdx.x] = 0.0f;
}

// ---------------- kernel 1: sum of squares of z (for sigma) ----------------
__global__ void lamp_sumsq(const float* __restrict__ zr,
                           const float* __restrict__ zi,
                           float* __restrict__ ws) {
  __shared__ float s;
  if (threadIdx.x == 0) s = 0.0f;
  __syncthreads();
  const int n4 = BATCH * KD / 4;
  const float4* a = (const float4*)zr;
  const float4* b = (const float4*)zi;
  float loc = 0.0f;
  for (int i = blockIdx.x * blockDim.x + threadIdx.x; i < n4;
       i += gridDim.x * blockDim.x) {
    float4 x = a[i];
    loc += x.x * x.x + x.y * x.y + x.z * x.z + x.w * x.w;
    float4 y = b[i];
    loc += y.x * y.x + y.y * y.y + y.z * y.z + y.w * y.w;
  }
  atomicAdd(&s, loc);
  __syncthreads();
  if (threadIdx.x == 0) atomicAdd(&ws[0], s);
}

// ---------------- kernel 2: GEMM1 + shrink + GEMM2 (fused) ----------------
// One block = 16 batch rows. 8 waves; each wave owns 4 column tiles of 16
// for GEMM1, then a K-chunk of 64 for GEMM2.
__global__ void __launch_bounds__(256)
lamp_main(const float* __restrict__ z_re, const float* __restrict__ z_im,
          const float* __restrict__ Hre, const float* __restrict__ Him,
          const float* __restrict__ Bre, const float* __restrict__ Bim,
          const float* __restrict__ Are, const float* __restrict__ Aim,
          const float* __restrict__ theta,
          float* __restrict__ out, float* __restrict__ ws) {
  extern __shared__ float smem[];
  float* sHr  = smem;               // 16 * 514
  float* sHi  = sHr + 16 * 514;     // 16 * 514
  float* sAcc = sHi + 16 * 514;     // 768 re + 768 im
  float* sDs  = sAcc + 1536;        // 2 scalars

  const int tid = threadIdx.x;
  for (int i = tid; i < 1538; i += 256) sAcc[i] = 0.0f;
  __syncthreads();

  const int lane = tid & 31;
  const int w    = tid >> 5;
  const int l16  = lane & 15;
  const int hi   = lane >> 4;
  const int m0   = blockIdx.x * 16;

  const float th0 = theta[0], th1 = theta[1], th2 = theta[2];
  const float sigma  = sqrtf(ws[0]) * 0.14433756729740643f;  // 1/sqrt(48)
  const float ts     = th0 * sigma;
  const float s2     = ts * ts;
  const float invs2  = 1.0f / s2;
  const float inv2s2 = 0.5f * invs2;

  // Hoist z fragments: a[v] = z[m0+l16, 4s + v + 2*hi] -> contiguous float2.
  v2f zr[12], zi2[12], nzi[12];
  {
    const int base = (m0 + l16) * KD + 2 * hi;
    #pragma unroll
    for (int s = 0; s < 12; ++s) {
      zr[s]  = *(const v2f*)(z_re + base + 4 * s);
      zi2[s] = *(const v2f*)(z_im + base + 4 * s);
      nzi[s] = -zi2[s];
    }
  }

  float* outHr = out + (long)2 * BATCH * KD;
  float* outHi = outHr + (long)BATCH * NN;

  float dsr = 0.0f, dsi = 0.0f;

  // ---- Phase A: Z = z @ B^T (complex), R = H_hat + Z, Hh = shrink(R) ----
  #pragma unroll
  for (int t = 0; t < 4; ++t) {
    const int n0 = (w * 4 + t) * 16;
    v8f cre = {0, 0, 0, 0, 0, 0, 0, 0};
    v8f cim = {0, 0, 0, 0, 0, 0, 0, 0};
    // b[v] = B^T[k, n0+l16] = B[n0+l16, k] -> contiguous float2 per lane.
    const int brow = (n0 + l16) * KD + 2 * hi;
    #pragma unroll
    for (int s = 0; s < 12; ++s) {
      v2f br = *(const v2f*)(Bre + brow + 4 * s);
      v2f bi = *(const v2f*)(Bim + brow + 4 * s);
      cre = __builtin_amdgcn_wmma_f32_16x16x4_f32(false, zr[s],  false, br,
                                                  (short)0, cre, false, false);
      cre = __builtin_amdgcn_wmma_f32_16x16x4_f32(false, nzi[s], false, bi,
                                                  (short)0, cre, false, false);
      cim = __builtin_amdgcn_wmma_f32_16x16x4_f32(false, zr[s],  false, bi,
                                                  (short)0, cim, false, false);
      cim = __builtin_amdgcn_wmma_f32_16x16x4_f32(false, zi2[s], false, br,
                                                  (short)0, cim, false, false);
    }
    // C/D layout: vgpr v holds row M = v + 8*hi, col = l16.
    const int rb = (m0 + 8 * hi) * NN + n0 + l16;
    #pragma unroll
    for (int v = 0; v < 8; ++v) {
      const int gidx = rb + v * NN;
      float Rr  = Hre[gidx] + cre[v];
      float Ri  = Him[gidx] + cim[v];
      float r2r = Rr * Rr, r2i = Ri * Ri;
      float er  = __expf(-r2r * inv2s2);
      float ei  = __expf(-r2i * inv2s2);
      float hr  = Rr * (th1 + th2 * er);
      float hmi = Ri * (th1 + th2 * ei);
      dsr += th1 + th2 * er * (1.0f - r2r * invs2);
      dsi += th1 + th2 * ei * (1.0f - r2i * invs2);
      outHr[gidx] = hr;
      outHi[gidx] = hmi;
      const int ml = v + 8 * hi;
      sHr[ml * 514 + n0 + l16] = hr;
      sHi[ml * 514 + n0 + l16] = hmi;
    }
  }

  atomicAdd(&sDs[0], dsr);
  atomicAdd(&sDs[1], dsi);
  __syncthreads();
  if (tid == 0) {
    atomicAdd(&ws[1], sDs[0]);
    atomicAdd(&ws[2], sDs[1]);
  }

  // ---- Phase B: h = Hh @ A (complex). K=512 split 64 per wave. ----
  v8f hreA[3], himA[3];
  #pragma unroll
  for (int nt = 0; nt < 3; ++nt) {
    hreA[nt] = (v8f){0, 0, 0, 0, 0, 0, 0, 0};
    himA[nt] = (v8f){0, 0, 0, 0, 0, 0, 0, 0};
  }
  const int kbase = w * 64;
  for (int s = 0; s < 16; ++s) {
    const int kk = kbase + 4 * s + 2 * hi;
    v2f hrF = *(const v2f*)(sHr + l16 * 514 + kk);
    v2f hiF = *(const v2f*)(sHi + l16 * 514 + kk);
    v2f nhiF = -hiF;
    #pragma unroll
    for (int nt = 0; nt < 3; ++nt) {
      const int c0 = nt * 16 + l16;
      const int r0 = kk * KD + c0;  // b[v] = A[kk+v, c0]
      v2f arF, aiF;
      arF.x = Are[r0];      arF.y = Are[r0 + KD];
      aiF.x = Aim[r0];      aiF.y = Aim[r0 + KD];
      hreA[nt] = __builtin_amdgcn_wmma_f32_16x16x4_f32(false, hrF,  false, arF,
                                    (short)0, hreA[nt], false, false);
      hreA[nt] = __builtin_amdgcn_wmma_f32_16x16x4_f32(false, nhiF, false, aiF,
                                    (short)0, hreA[nt], false, false);
      himA[nt] = __builtin_amdgcn_wmma_f32_16x16x4_f32(false, hrF,  false, aiF,
                                    (short)0, himA[nt], false, false);
      himA[nt] = __builtin_amdgcn_wmma_f32_16x16x4_f32(false, hiF,  false, arF,
                                    (short)0, himA[nt], false, false);
    }
  }

  // Cross-wave reduction of partial h into LDS.
  #pragma unroll
  for (int nt = 0; nt < 3; ++nt) {
    #pragma unroll
    for (int v = 0; v < 8; ++v) {
      const int ml = v + 8 * hi;
      const int c  = nt * 16 + l16;
      atomicAdd(&sAcc[ml * KD + c],       hreA[nt][v]);
      atomicAdd(&sAcc[768 + ml * KD + c], himA[nt][v]);
    }
  }
  __syncthreads();

  // Stash h into the d_out z slots (finalize kernel rewrites them).
  float* outZr = out;
  float* outZi = out + (long)BATCH * KD;
  for (int i = tid; i < 768; i += 256) {
    const int ml = i / KD, c = i % KD;
    const long g = (long)(m0 + ml) * KD + c;
    outZr[g] = sAcc[i];
    outZi[g] = sAcc[768 + i];
  }
}

// ---------------- kernel 3: z_new = u - h + b*z (in place over h) ----------
__global__ void lamp_finalize(const float* __restrict__ u_re,
                              const float* __restrict__ u_im,
                              const float* __restrict__ z_re,
                              const float* __restrict__ z_im,
                              const float* __restrict__ ws,
                              float* __restrict__ out) {
  const int i = blockIdx.x * 256 + threadIdx.x;
  if (i >= BATCH * KD) return;
  const float bre = ws[1] * (1.0f / KD);
  const float bim = ws[2] * (1.0f / KD);
  const float hr = out[i];
  out[i] = u_re[i] - hr + bre * z_re[i];
  const long j = (long)BATCH * KD + i;
  const float hv = out[j];
  out[j] = u_im[i] - hv + bim * z_im[i];
}

extern "C" void kernel_launch(void* const* d_in, const int* in_sizes, int n_in,
                              void* d_out, int out_size, void* d_ws,
                              size_t ws_size, hipStream_t stream) {
  const float* u_re = (const float*)d_in[0];
  const float* u_im = (const float*)d_in[1];
  const float* z_re = (const float*)d_in[2];
  const float* z_im = (const float*)d_in[3];
  const float* Hre  = (const float*)d_in[4];
  const float* Him  = (const float*)d_in[5];
  const float* Bre  = (const float*)d_in[6];
  const float* Bim  = (const float*)d_in[7];
  const float* Are  = (const float*)d_in[8];
  const float* Aim  = (const float*)d_in[9];
  const float* theta = (const float*)d_in[10];
  float* out = (float*)d_out;
  float* ws  = (float*)d_ws;

  lamp_init_ws<<<1, 32, 0, stream>>>(ws);
  lamp_sumsq<<<1024, 256, 0, stream>>>(z_re, z_im, ws);

  const size_t shmem = (size_t)(2 * 16 * 514 + 1536 + 2) * sizeof(float);
  lamp_main<<<BATCH / 16, 256, shmem, stream>>>(z_re, z_im, Hre, Him, Bre, Bim,
                                                Are, Aim, theta, out, ws);

  lamp_finalize<<<(BATCH * KD + 255) / 256, 256, 0, stream>>>(u_re, u_im, z_re,
                                                              z_im, ws, out);
}